// Head_32366873543380
// MI455X (gfx1250) — compile-verified
//
#include <hip/hip_runtime.h>
#include <hip/hip_bf16.h>
#include <math.h>

// Problem shape (fixed by the reference)
#define NB 8
#define NT 2048
#define NE 1024
#define NH 128

typedef __attribute__((ext_vector_type(2))) float v2f;
typedef __attribute__((ext_vector_type(8))) float v8f;

// Full-precision fp32 matrix op: V_WMMA_F32_16X16X4_F32
// (neg_a, A, neg_b, B, c_mod, C, reuse_a, reuse_b) -- codegen-verified round 1
__device__ __forceinline__ v8f wmma_f32x4(v2f a, v2f b, v8f c) {
  return __builtin_amdgcn_wmma_f32_16x16x4_f32(false, a, false, b, (short)0, c,
                                               false, false);
}

// ---------------------------------------------------------------------------
// Kernel 1: q/k/v = x @ W{q,k,v}.  x:(NB*NT, NE) row-major, W:(NE, NH).
// One 16-row tile per block; 8 waves cover the 8 16-col tiles of NH=128.
// One A-fragment load feeds three WMMAs (shared across Wq/Wk/Wv).
// ---------------------------------------------------------------------------
__global__ __launch_bounds__(256) void qkv_kernel(
    const float* __restrict__ x, const float* __restrict__ Wq,
    const float* __restrict__ Wk, const float* __restrict__ Wv,
    float* __restrict__ qo, float* __restrict__ ko, float* __restrict__ vo) {
  const int lane = threadIdx.x & 31;
  const int wid  = threadIdx.x >> 5;        // 0..7 -> N tile
  const int lh   = lane & 15;               // M row (A) / N col (B,C)
  const int khi  = (lane >> 4) << 1;        // K offset 0 or 2
  const int row0 = blockIdx.x * 16;
  const int col0 = wid * 16;

  const float* xr = x + (size_t)(row0 + lh) * NE;

  v8f aq = {}, ak = {}, av = {};
  for (int kk = 0; kk < NE; kk += 4) {
    v2f a = *(const v2f*)(xr + kk + khi);   // A[lh][kk+khi .. +1]
    const float* pq = Wq + (size_t)(kk + khi) * NH + col0 + lh;
    const float* pk = Wk + (size_t)(kk + khi) * NH + col0 + lh;
    const float* pv = Wv + (size_t)(kk + khi) * NH + col0 + lh;
    v2f bq; bq.x = pq[0]; bq.y = pq[NH];    // B[khi][lh], B[khi+1][lh]
    v2f bk; bk.x = pk[0]; bk.y = pk[NH];
    v2f bv; bv.x = pv[0]; bv.y = pv[NH];
    aq = wmma_f32x4(a, bq, aq);
    ak = wmma_f32x4(a, bk, ak);
    av = wmma_f32x4(a, bv, av);
  }
  // D layout: vgpr r, lanes 0-15 -> row r, lanes 16-31 -> row r+8, col = lh
  const int rbase = row0 + ((lane >> 4) << 3);
  for (int r = 0; r < 8; ++r) {
    size_t idx = (size_t)(rbase + r) * NH + col0 + lh;
    qo[idx] = aq[r]; ko[idx] = ak[r]; vo[idx] = av[r];
  }
}

// ---------------------------------------------------------------------------
// Kernel 2: flash-style causal attention, one wave per 16-query block.
// Q frags + O accumulator live in VGPRs; scores never touch memory.
// P (D-layout) -> A-layout conversion bounces through a 1KB/wave LDS tile.
// Wave->qblk mapping is pair-balanced: idx -> (idx&1 ? 127-idx/2 : idx/2),
// so every (even,odd) wave pair does (qblk+1)+(128-qblk) = 129 key blocks:
// with ~1 wave per SIMD the critical path is the slowest wave, and this
// flattens it from 128 units to ~65.
// ---------------------------------------------------------------------------
__global__ __launch_bounds__(256) void attn_kernel(
    const float* __restrict__ q, const float* __restrict__ k,
    const float* __restrict__ v, float* __restrict__ out) {
  __shared__ float pbuf[8][16 * 16];
  const int lane = threadIdx.x & 31;
  const int wid  = threadIdx.x >> 5;
  const int lh   = lane & 15;
  const int hi   = lane >> 4;               // 0/1 half
  const int khi  = hi << 1;

  const int gw   = blockIdx.x * 8 + wid;    // 0 .. NB*(NT/16)-1
  const int b    = gw >> 7;                 // / (NT/16)
  const int widx = gw & 127;
  const int qblk = (widx & 1) ? (127 - (widx >> 1)) : (widx >> 1);

  const float* qb = q + ((size_t)b * NT + (size_t)qblk * 16) * NH;
  const float* kb = k + (size_t)b * NT * NH;
  const float* vb = v + (size_t)b * NT * NH;

  // Preload Q A-fragments: 16x128 fp32 in 32 v2f (64 VGPRs)
  v2f qf[32];
  {
    const float* qr = qb + (size_t)lh * NH;
    for (int i = 0; i < 32; ++i) qf[i] = *(const v2f*)(qr + i * 4 + khi);
  }

  v8f o[8];
  for (int nt = 0; nt < 8; ++nt) o[nt] = (v8f){};
  float m[8], l[8];
  for (int r = 0; r < 8; ++r) { m[r] = -INFINITY; l[r] = 0.f; }

  const float scale = sqrtf((float)NT);     // reference multiplies by sqrt(T)
  float* pl = &pbuf[wid][0];

  for (int j = 0; j <= qblk; ++j) {
    // ---- S = Q @ K_j^T (16x16, fp32 WMMA over K=128) ----
    const float* kr = kb + ((size_t)j * 16 + lh) * NH;
    v8f s = {};
    for (int i = 0; i < 32; ++i) {
      v2f bf = *(const v2f*)(kr + i * 4 + khi);
      s = wmma_f32x4(qf[i], bf, s);
    }
    // prefetch next K/V tiles while we do softmax math
    if (j < qblk) {
      __builtin_prefetch(kr + 16 * NH, 0, 0);
      __builtin_prefetch(vb + ((size_t)(j + 1) * 16 + lh) * NH, 0, 0);
    }
    // ---- scale + causal mask (only the diagonal block needs masking) ----
    if (j == qblk) {
      for (int r = 0; r < 8; ++r) {
        int qrow = r + (hi << 3);
        s[r] = (lh > qrow) ? -INFINITY : s[r] * scale;
      }
    } else {
      for (int r = 0; r < 8; ++r) s[r] *= scale;
    }
    // ---- online softmax: row stats via half-wave xor reductions ----
    float c[8];
    for (int r = 0; r < 8; ++r) {
      float mx = s[r];
      mx = fmaxf(mx, __shfl_xor(mx, 1));
      mx = fmaxf(mx, __shfl_xor(mx, 2));
      mx = fmaxf(mx, __shfl_xor(mx, 4));
      mx = fmaxf(mx, __shfl_xor(mx, 8));
      float mnew = fmaxf(m[r], mx);
      float corr = __expf(m[r] - mnew);     // first block: exp(-inf)=0
      float p    = __expf(s[r] - mnew);     // masked: exp(-inf)=0
      float rs = p;
      rs += __shfl_xor(rs, 1);
      rs += __shfl_xor(rs, 2);
      rs += __shfl_xor(rs, 4);
      rs += __shfl_xor(rs, 8);
      l[r] = l[r] * corr + rs;
      m[r] = mnew;
      c[r] = corr;
      s[r] = p;                             // s now holds P
    }
    for (int nt = 0; nt < 8; ++nt)
      for (int r = 0; r < 8; ++r) o[nt][r] *= c[r];
    // ---- P: D-layout -> A-layout via LDS (wave-private, DS in-order) ----
    for (int r = 0; r < 8; ++r)
      pl[(r + (hi << 3)) * 16 + lh] = s[r];
    v2f pf[4];
    for (int kc = 0; kc < 4; ++kc)
      pf[kc] = *(const v2f*)(pl + lh * 16 + kc * 4 + khi);
    // ---- O += P @ V_j (16x16 @ 16x128) ----
    const float* vr = vb + ((size_t)j * 16 + khi) * NH + lh;
    for (int nt = 0; nt < 8; ++nt) {
      for (int kc = 0; kc < 4; ++kc) {
        v2f bv;
        bv.x = vr[(kc * 4 + 0) * NH + nt * 16];
        bv.y = vr[(kc * 4 + 1) * NH + nt * 16];
        o[nt] = wmma_f32x4(pf[kc], bv, o[nt]);
      }
    }
  }
  // ---- epilogue: divide by row sums, store (B,T,H) fp32 ----
  const int qrow0 = qblk * 16 + (hi << 3);
  float rl[8];
  for (int r = 0; r < 8; ++r) rl[r] = 1.0f / l[r];
  for (int nt = 0; nt < 8; ++nt)
    for (int r = 0; r < 8; ++r)
      out[((size_t)b * NT + qrow0 + r) * NH + nt * 16 + lh] =
          o[nt][r] * rl[r];
}

// ---------------------------------------------------------------------------
extern "C" void kernel_launch(void* const* d_in, const int* in_sizes, int n_in,
                              void* d_out, int out_size, void* d_ws,
                              size_t ws_size, hipStream_t stream) {
  const float* x  = (const float*)d_in[0];
  const float* Wq = (const float*)d_in[1];
  const float* Wk = (const float*)d_in[2];
  const float* Wv = (const float*)d_in[3];

  const size_t qkvElems = (size_t)NB * NT * NH;   // 2M floats each
  float* qo = (float*)d_ws;
  float* ko = qo + qkvElems;
  float* vo = ko + qkvElems;

  // Stage 1: QKV projection. 1024 M-tiles, 8 waves/block cover NH.
  qkv_kernel<<<(NB * NT) / 16, 256, 0, stream>>>(x, Wq, Wk, Wv, qo, ko, vo);
  // Stage 2: flash attention. 1024 query blocks, 8 (pair-balanced) waves per
  // workgroup.
  attn_kernel<<<(NB * NT / 16) / 8, 256, 0, stream>>>(qo, ko, vo,
                                                      (float*)d_out);
}